// VAE_mat_1ring_19524921328204
// MI455X (gfx1250) — compile-verified
//
#include <hip/hip_runtime.h>
#include <hip/hip_bf16.h>

// ---------------------------------------------------------------------------
// Spherical-mesh VAE forward for MI455X (gfx1250, wave32, WMMA).
// Every matmul is D = A(16xK-tile) x B(Kx16) + C via V_WMMA_F32_16X16X4_F32.
// Layouts (ISA 7.12.2, 32-bit):
//   A 16x4 : a.x = A[m][k0+2*hi], a.y = A[m][k0+2*hi+1]   (m=lane&15, hi=lane>>4)
//   B 4x16 : b.x = B[k0+2*hi][n], b.y = B[k0+2*hi+1][n]   (n=lane&15)
//   C/D    : vgpr v, lanes0-15 -> M=v, lanes16-31 -> M=v+8, N=lane&15
// ---------------------------------------------------------------------------

typedef float v2f __attribute__((ext_vector_type(2)));
typedef float v8f __attribute__((ext_vector_type(8)));

__device__ __forceinline__ v8f wmma4(v2f a, v2f b, v8f c) {
  return __builtin_amdgcn_wmma_f32_16x16x4_f32(false, a, false, b, (short)0, c,
                                               false, false);
}

// ------------------------- pool: mean over 7 neighbors ----------------------
__global__ __launch_bounds__(256) void pool_mean_kernel(
    const float* __restrict__ x, const int* __restrict__ neigh,
    float* __restrict__ out, int nc, int ci) {
  long long total = (long long)nc * ci * 16;
  for (long long e = (long long)blockIdx.x * blockDim.x + threadIdx.x;
       e < total; e += (long long)gridDim.x * blockDim.x) {
    int b = (int)(e & 15);
    long long t = e >> 4;
    int c = (int)(t % ci);
    int node = (int)(t / ci);
    float s = 0.f;
#pragma unroll
    for (int j = 0; j < 7; ++j) {
      int nb = neigh[node * 7 + j];
      s += x[((size_t)nb * ci + c) * 16 + b];
    }
    out[e] = s * (1.0f / 7.0f);
  }
}

// ------------------- conv1ring: out[n] = W @ gather7(x)[n] ------------------
// One wave handles (node pair, m-tile of 16 output chans); A loads shared by
// the two per-node WMMAs.  K = 7*ci, ci % 4 == 0 or ci == 4 (both hold here),
// so a K-step of 4 never crosses a neighbor block.
__global__ __launch_bounds__(256) void conv1ring_wmma(
    const float* __restrict__ x, const int* __restrict__ neigh,
    const float* __restrict__ W, const float* __restrict__ bias,
    float* __restrict__ out, int n, int ci, int co) {
  const int K = 7 * ci;
  const int mt = (co + 15) >> 4;
  const long long units = (long long)(n >> 1) * mt;
  const int lane = threadIdx.x & 31;
  const int low = lane & 15, hi = lane >> 4;

  long long unit = (long long)blockIdx.x * 8 + (threadIdx.x >> 5);
  const long long ustride = (long long)gridDim.x * 8;
  for (; unit < units; unit += ustride) {
    const int m = (int)(unit % mt);
    const int np = (int)(unit / mt);
    const int node0 = np * 2, node1 = node0 + 1;
    const int mbase = m << 4;
    int arow = mbase + low;
    if (arow >= co) arow = co - 1;  // clamp (results discarded on store)
    const float* Wp = W + (size_t)arow * K + 2 * hi;

    v8f acc0 = {0.f, 0.f, 0.f, 0.f, 0.f, 0.f, 0.f, 0.f};
    v8f acc1 = {0.f, 0.f, 0.f, 0.f, 0.f, 0.f, 0.f, 0.f};

    for (int j = 0; j < 7; ++j) {
      const int nb0 = neigh[node0 * 7 + j];
      const int nb1 = neigh[node1 * 7 + j];
      const float* xb0 = x + ((size_t)nb0 * ci + 2 * hi) * 16 + low;
      const float* xb1 = x + ((size_t)nb1 * ci + 2 * hi) * 16 + low;
      const float* wj = Wp + j * ci;
      for (int c = 0; c < ci; c += 4) {
        v2f a = *(const v2f*)(wj + c);
        v2f b0, b1;
        b0.x = xb0[c * 16];
        b0.y = xb0[c * 16 + 16];
        b1.x = xb1[c * 16];
        b1.y = xb1[c * 16 + 16];
        acc0 = wmma4(a, b0, acc0);
        acc1 = wmma4(a, b1, acc1);
      }
    }
#pragma unroll
    for (int v = 0; v < 8; ++v) {
      const int mrow = mbase + v + 8 * hi;
      if (mrow < co) {
        const float bv = bias[mrow];
        out[((size_t)node0 * co + mrow) * 16 + low] = acc0[v] + bv;
        out[((size_t)node1 * co + mrow) * 16 + low] = acc1[v] + bv;
      }
    }
  }
}

// ------------- dense per-node GEMM: out[n] = A(MxK) @ B[n](Kx16) ------------
// Used for upconv (per-node) and the w3 decode GEMM (n == 1).
__global__ __launch_bounds__(256) void dense_pernode_wmma(
    const float* __restrict__ A, const float* __restrict__ bias,
    const float* __restrict__ B, float* __restrict__ out, int n, int M,
    int K) {
  const int mt = (M + 15) >> 4;
  const long long units = (long long)n * mt;
  const int lane = threadIdx.x & 31;
  const int low = lane & 15, hi = lane >> 4;

  long long unit = (long long)blockIdx.x * 8 + (threadIdx.x >> 5);
  const long long ustride = (long long)gridDim.x * 8;
  for (; unit < units; unit += ustride) {
    const int m = (int)(unit % mt);
    const int node = (int)(unit / mt);
    const int mbase = m << 4;
    int arow = mbase + low;
    if (arow >= M) arow = M - 1;
    const float* Ap = A + (size_t)arow * K + 2 * hi;
    const float* Bp = B + ((size_t)node * K + 2 * hi) * 16 + low;

    v8f acc = {0.f, 0.f, 0.f, 0.f, 0.f, 0.f, 0.f, 0.f};
    for (int k = 0; k < K; k += 4) {
      v2f a = *(const v2f*)(Ap + k);
      v2f b;
      b.x = Bp[k * 16];
      b.y = Bp[k * 16 + 16];
      acc = wmma4(a, b, acc);
    }
#pragma unroll
    for (int v = 0; v < 8; ++v) {
      const int mrow = mbase + v + 8 * hi;
      if (mrow < M)
        out[((size_t)node * M + mrow) * 16 + low] = acc[v] + bias[mrow];
    }
  }
}

// --------- K-split GEMM (n==1) for mu/logvar: deterministic partials --------
__global__ __launch_bounds__(256) void dense_ksplit_wmma(
    const float* __restrict__ A, const float* __restrict__ B,
    float* __restrict__ pout, int M, int K, int kchunks) {
  const int mt = M >> 4;  // M multiple of 16
  const long long units = (long long)mt * kchunks;
  const int lane = threadIdx.x & 31;
  const int low = lane & 15, hi = lane >> 4;
  const int klen = K / kchunks;  // multiple of 4

  long long unit = (long long)blockIdx.x * 8 + (threadIdx.x >> 5);
  const long long ustride = (long long)gridDim.x * 8;
  for (; unit < units; unit += ustride) {
    const int kc = (int)(unit % kchunks);
    const int m = (int)(unit / kchunks);
    const int mbase = m << 4;
    const int arow = mbase + low;
    const int k0base = kc * klen;
    const float* Ap = A + (size_t)arow * K + k0base + 2 * hi;
    const float* Bp = B + ((size_t)k0base + 2 * hi) * 16 + low;

    v8f acc = {0.f, 0.f, 0.f, 0.f, 0.f, 0.f, 0.f, 0.f};
    for (int k = 0; k < klen; k += 4) {
      v2f a = *(const v2f*)(Ap + k);
      v2f b;
      b.x = Bp[k * 16];
      b.y = Bp[k * 16 + 16];
      acc = wmma4(a, b, acc);
    }
#pragma unroll
    for (int v = 0; v < 8; ++v) {
      const int mrow = mbase + v + 8 * hi;
      pout[((size_t)kc * M + mrow) * 16 + low] = acc[v];
    }
  }
}

__global__ __launch_bounds__(256) void ksplit_reduce_kernel(
    const float* __restrict__ pout, const float* __restrict__ bias,
    float* __restrict__ out, int M, int kchunks) {
  int e = blockIdx.x * blockDim.x + threadIdx.x;
  if (e < M * 16) {
    float s = bias[e >> 4];
    for (int kc = 0; kc < kchunks; ++kc) s += pout[(size_t)kc * M * 16 + e];
    out[e] = s;
  }
}

// ---------------- batchnorm: per-channel stats over (nodes x 16) ------------
__global__ __launch_bounds__(256) void bn_stats_kernel(
    const float* __restrict__ xin, float* __restrict__ mean,
    float* __restrict__ var, int n, int C) {
  const int c = blockIdx.x;
  const int total = n * 16;
  float s = 0.f, sq = 0.f;
  for (int t = threadIdx.x; t < total; t += 256) {
    const float v = xin[((size_t)(t >> 4) * C + c) * 16 + (t & 15)];
    s += v;
    sq += v * v;
  }
  __shared__ float shs[256];
  __shared__ float shq[256];
  shs[threadIdx.x] = s;
  shq[threadIdx.x] = sq;
  __syncthreads();
  for (int off = 128; off > 0; off >>= 1) {
    if ((int)threadIdx.x < off) {
      shs[threadIdx.x] += shs[threadIdx.x + off];
      shq[threadIdx.x] += shq[threadIdx.x + off];
    }
    __syncthreads();
  }
  if (threadIdx.x == 0) {
    const float m = shs[0] / (float)total;
    mean[c] = m;
    var[c] = shq[0] / (float)total - m * m;
  }
}

__global__ __launch_bounds__(256) void bn_lrelu_kernel(
    float* __restrict__ xio, const float* __restrict__ g,
    const float* __restrict__ be, const float* __restrict__ mean,
    const float* __restrict__ var, long long total, int C) {
  for (long long e = (long long)blockIdx.x * blockDim.x + threadIdx.x;
       e < total; e += (long long)gridDim.x * blockDim.x) {
    const int c = (int)((e >> 4) % C);
    const float y =
        g[c] * (xio[e] - mean[c]) * rsqrtf(var[c] + 1e-5f) + be[c];
    xio[e] = (y >= 0.f) ? y : 0.2f * y;
  }
}

// ---------------- upconv scatter: top rows + mean of down pairs -------------
__global__ __launch_bounds__(256) void upconv_gather_kernel(
    const float* __restrict__ yf, const int* __restrict__ top,
    const int* __restrict__ down, float* __restrict__ out, int nc, int nn,
    int co) {
  const int rs = co * 16;
  long long total = (long long)nn * rs;
  for (long long e = (long long)blockIdx.x * blockDim.x + threadIdx.x;
       e < total; e += (long long)gridDim.x * blockDim.x) {
    const int r = (int)(e % rs);
    const int node = (int)(e / rs);
    if (node < nc) {
      out[e] = yf[(size_t)top[node] * rs + r];
    } else {
      const int k = node - nc;
      const int i0 = down[2 * k], i1 = down[2 * k + 1];
      out[e] = 0.5f * (yf[(size_t)i0 * rs + r] + yf[(size_t)i1 * rs + r]);
    }
  }
}

// -------- latent: z = mu + eps*exp(0.5*logvar); also emit mu/logvar --------
__global__ __launch_bounds__(256) void latent_kernel(
    const float* __restrict__ muT, const float* __restrict__ lvT,
    const float* __restrict__ eps, float* __restrict__ zT,
    float* __restrict__ mu_out, float* __restrict__ lv_out) {
  int e = blockIdx.x * blockDim.x + threadIdx.x;  // over 256*16
  if (e < 256 * 16) {
    const int l = e >> 4, b = e & 15;
    const float m = muT[e], lv = lvT[e];
    mu_out[b * 256 + l] = m;
    lv_out[b * 256 + l] = lv;
    zT[e] = m + eps[b * 256 + l] * expf(0.5f * lv);
  }
}

__global__ __launch_bounds__(256) void sigmoid_kernel(
    const float* __restrict__ in, float* __restrict__ out, long long total) {
  for (long long e = (long long)blockIdx.x * blockDim.x + threadIdx.x;
       e < total; e += (long long)gridDim.x * blockDim.x) {
    out[e] = 1.0f / (1.0f + expf(-in[e]));
  }
}

// ---------------------------------------------------------------------------
// Host orchestration
// ---------------------------------------------------------------------------
static inline int ew_blocks(long long total) {
  long long b = (total + 255) / 256;
  if (b > 65535) b = 65535;
  if (b < 1) b = 1;
  return (int)b;
}

extern "C" void kernel_launch(void* const* d_in, const int* in_sizes, int n_in,
                              void* d_out, int out_size, void* d_ws,
                              size_t ws_size, hipStream_t stream) {
  (void)in_sizes; (void)n_in; (void)out_size; (void)ws_size;

  // ---- input pointer map (setup_inputs dict order, flattened) ----
  const float* X = (const float*)d_in[0];    // (40962,4,16)
  const float* EPS = (const float*)d_in[1];  // (16,256)
  auto encp = [&](int i, int k) { return (const float*)d_in[2 + 8 * i + k]; };
  // enc keys: 0=w1 1=b1 2=g1 3=be1 4=w2 5=b2 6=g2 7=be2
  auto decp = [&](int j, int k) { return (const float*)d_in[42 + 10 * j + k]; };
  // dec keys: 0=uw 1=ub 2=w1 3=b1 4=g1 5=be1 6=w2 7=b2 8=g2 9=be2
  const float* FW1 = (const float*)d_in[92];
  const float* FB1 = (const float*)d_in[93];
  const float* FW2 = (const float*)d_in[94];
  const float* FB2 = (const float*)d_in[95];
  const float* FW3 = (const float*)d_in[96];
  const float* FB3 = (const float*)d_in[97];
  const int* NEI[6];
  for (int i = 0; i < 6; ++i) NEI[i] = (const int*)d_in[98 + i];
  const int* TOPI[5];
  for (int j = 0; j < 5; ++j) TOPI[j] = (const int*)d_in[104 + j];
  const int* DWNI[5];
  for (int j = 0; j < 5; ++j) DWNI[j] = (const int*)d_in[109 + j];

  // ---- workspace carve (floats) ----
  float* ws = (float*)d_ws;
  const size_t SZ_AB = (size_t)10242 * 64 * 16;       // 10,487,808
  const size_t SZ_C = (size_t)2562 * 448 * 16;        // 18,364,416
  float* A = ws;
  float* Bb = ws + SZ_AB;
  float* Cc = ws + 2 * SZ_AB;
  float* small = ws + 2 * SZ_AB + SZ_C;
  float* mean = small;           // 512
  float* var = small + 512;      // 512
  float* muT = small + 1024;     // 4096  (256 x 16)
  float* lvT = small + 5120;     // 4096
  float* zT = small + 9216;      // 4096
  float* pks = small + 13312;    // 32*256*16 = 131072

  float* OUT_Y = (float*)d_out;                 // (40962,4,16)
  float* OUT_MU = (float*)d_out + 2621568;      // (16,256)
  float* OUT_LV = (float*)d_out + 2621568 + 4096;

  // ---- launch helpers ----
  auto pool = [&](const float* xin, const int* ne, float* out, int nc, int ci) {
    pool_mean_kernel<<<ew_blocks((long long)nc * ci * 16), 256, 0, stream>>>(
        xin, ne, out, nc, ci);
  };
  auto conv = [&](const float* xin, const int* ne, const float* W,
                  const float* bb, float* out, int n, int ci, int co) {
    const int mt = (co + 15) >> 4;
    long long units = (long long)(n / 2) * mt;
    long long blocks = (units + 7) / 8;
    if (blocks > 32768) blocks = 32768;
    if (blocks < 1) blocks = 1;
    conv1ring_wmma<<<(int)blocks, 256, 0, stream>>>(xin, ne, W, bb, out, n, ci,
                                                    co);
  };
  auto bn = [&](float* buf, const float* g, const float* be, int n, int C) {
    bn_stats_kernel<<<C, 256, 0, stream>>>(buf, mean, var, n, C);
    long long total = (long long)n * C * 16;
    bn_lrelu_kernel<<<ew_blocks(total), 256, 0, stream>>>(buf, g, be, mean,
                                                          var, total, C);
  };
  auto dense = [&](const float* Am, const float* bb, const float* Bm,
                   float* out, int n, int M, int K) {
    const int mt = (M + 15) >> 4;
    long long units = (long long)n * mt;
    long long blocks = (units + 7) / 8;
    if (blocks > 32768) blocks = 32768;
    if (blocks < 1) blocks = 1;
    dense_pernode_wmma<<<(int)blocks, 256, 0, stream>>>(Am, bb, Bm, out, n, M,
                                                        K);
  };
  auto gather = [&](const float* yf, int j, float* out, int nc, int nn,
                    int co) {
    upconv_gather_kernel<<<ew_blocks((long long)nn * co * 16), 256, 0,
                           stream>>>(yf, TOPI[j], DWNI[j], out, nc, nn, co);
  };

  // ================= encoder =================
  // i=0: pool 40962->10242 (ci=4), conv 4->64, conv 64->64
  pool(X, NEI[0], A, 10242, 4);
  conv(A, NEI[1], encp(0, 0), encp(0, 1), Bb, 10242, 4, 64);
  bn(Bb, encp(0, 2), encp(0, 3), 10242, 64);
  conv(Bb, NEI[1], encp(0, 4), encp(0, 5), A, 10242, 64, 64);
  bn(A, encp(0, 6), encp(0, 7), 10242, 64);
  // i=1: 10242->2562, 64->64, 64->64
  pool(A, NEI[1], Bb, 2562, 64);
  conv(Bb, NEI[2], encp(1, 0), encp(1, 1), A, 2562, 64, 64);
  bn(A, encp(1, 2), encp(1, 3), 2562, 64);
  conv(A, NEI[2], encp(1, 4), encp(1, 5), Bb, 2562, 64, 64);
  bn(Bb, encp(1, 6), encp(1, 7), 2562, 64);
  // i=2: 2562->642, 64->128, 128->128
  pool(Bb, NEI[2], A, 642, 64);
  conv(A, NEI[3], encp(2, 0), encp(2, 1), Bb, 642, 64, 128);
  bn(Bb, encp(2, 2), encp(2, 3), 642, 128);
  conv(Bb, NEI[3], encp(2, 4), encp(2, 5), A, 642, 128, 128);
  bn(A, encp(2, 6), encp(2, 7), 642, 128);
  // i=3: 642->162, 128->256, 256->256
  pool(A, NEI[3], Bb, 162, 128);
  conv(Bb, NEI[4], encp(3, 0), encp(3, 1), A, 162, 128, 256);
  bn(A, encp(3, 2), encp(3, 3), 162, 256);
  conv(A, NEI[4], encp(3, 4), encp(3, 5), Bb, 162, 256, 256);
  bn(Bb, encp(3, 6), encp(3, 7), 162, 256);
  // i=4: 162->42, 256->512, 512->512
  pool(Bb, NEI[4], A, 42, 256);
  conv(A, NEI[5], encp(4, 0), encp(4, 1), Bb, 42, 256, 512);
  bn(Bb, encp(4, 2), encp(4, 3), 42, 512);
  conv(Bb, NEI[5], encp(4, 4), encp(4, 5), A, 42, 512, 512);
  bn(A, encp(4, 6), encp(4, 7), 42, 512);

  // ================= latent FCs =================
  // A holds h as (42,512,16) == B-matrix (21504 x 16) for WMMA directly.
  const int KFC = 42 * 512;  // 21504
  const int KCH = 32;        // 21504/32 = 672, multiple of 4
  {
    long long units = (long long)(256 / 16) * KCH;
    int blocks = (int)((units + 7) / 8);
    dense_ksplit_wmma<<<blocks, 256, 0, stream>>>(FW1, A, pks, 256, KFC, KCH);
    ksplit_reduce_kernel<<<(256 * 16 + 255) / 256, 256, 0, stream>>>(
        pks, FB1, muT, 256, KCH);
    dense_ksplit_wmma<<<blocks, 256, 0, stream>>>(FW2, A, pks, 256, KFC, KCH);
    ksplit_reduce_kernel<<<(256 * 16 + 255) / 256, 256, 0, stream>>>(
        pks, FB2, lvT, 256, KCH);
  }
  latent_kernel<<<(256 * 16 + 255) / 256, 256, 0, stream>>>(muT, lvT, EPS, zT,
                                                            OUT_MU, OUT_LV);
  // g = z @ w3.T + b3 -> y (42,512,16) in Bb
  dense(FW3, FB3, zT, Bb, 1, KFC, 256);

  // ================= decoder =================
  // j=0: 42 -> 162, ci=512 co=256
  dense(decp(0, 0), decp(0, 1), Bb, Cc, 42, 7 * 256, 512);
  gather(Cc, 0, A, 42, 162, 256);
  conv(A, NEI[4], decp(0, 2), decp(0, 3), Bb, 162, 256, 256);
  bn(Bb, decp(0, 4), decp(0, 5), 162, 256);
  conv(Bb, NEI[4], decp(0, 6), decp(0, 7), A, 162, 256, 256);
  bn(A, decp(0, 8), decp(0, 9), 162, 256);
  // j=1: 162 -> 642, ci=256 co=128
  dense(decp(1, 0), decp(1, 1), A, Cc, 162, 7 * 128, 256);
  gather(Cc, 1, Bb, 162, 642, 128);
  conv(Bb, NEI[3], decp(1, 2), decp(1, 3), A, 642, 128, 128);
  bn(A, decp(1, 4), decp(1, 5), 642, 128);
  conv(A, NEI[3], decp(1, 6), decp(1, 7), Bb, 642, 128, 128);
  bn(Bb, decp(1, 8), decp(1, 9), 642, 128);
  // j=2: 642 -> 2562, ci=128 co=64
  dense(decp(2, 0), decp(2, 1), Bb, Cc, 642, 7 * 64, 128);
  gather(Cc, 2, A, 642, 2562, 64);
  conv(A, NEI[2], decp(2, 2), decp(2, 3), Bb, 2562, 64, 64);
  bn(Bb, decp(2, 4), decp(2, 5), 2562, 64);
  conv(Bb, NEI[2], decp(2, 6), decp(2, 7), A, 2562, 64, 64);
  bn(A, decp(2, 8), decp(2, 9), 2562, 64);
  // j=3: 2562 -> 10242, ci=64 co=64
  dense(decp(3, 0), decp(3, 1), A, Cc, 2562, 7 * 64, 64);
  gather(Cc, 3, Bb, 2562, 10242, 64);
  conv(Bb, NEI[1], decp(3, 2), decp(3, 3), A, 10242, 64, 64);
  bn(A, decp(3, 4), decp(3, 5), 10242, 64);
  conv(A, NEI[1], decp(3, 6), decp(3, 7), Bb, 10242, 64, 64);
  bn(Bb, decp(3, 8), decp(3, 9), 10242, 64);
  // j=4: 10242 -> 40962, ci=64 co=4
  dense(decp(4, 0), decp(4, 1), Bb, Cc, 10242, 7 * 4, 64);
  gather(Cc, 4, A, 10242, 40962, 4);
  conv(A, NEI[0], decp(4, 2), decp(4, 3), Bb, 40962, 4, 4);
  bn(Bb, decp(4, 4), decp(4, 5), 40962, 4);
  conv(Bb, NEI[0], decp(4, 6), decp(4, 7), A, 40962, 4, 4);
  bn(A, decp(4, 8), decp(4, 9), 40962, 4);

  // final sigmoid -> d_out
  sigmoid_kernel<<<ew_blocks((long long)40962 * 4 * 16), 256, 0, stream>>>(
      A, OUT_Y, (long long)40962 * 4 * 16);
}